// ContMix_15719580304519
// MI455X (gfx1250) — compile-verified
//
#include <hip/hip_runtime.h>
#include <hip/hip_bf16.h>

// Problem constants (match reference)
constexpr int B  = 4;
constexpr int C  = 384;
constexpr int H  = 56;
constexpr int W  = 56;
constexpr int HW = H * W;          // 3136
constexpr int S  = 7;
constexpr int SS = S * S;          // 49
constexpr int SP = 64;             // SS padded to multiple of 16 for WMMA N-dim
constexpr int D  = C / 2;          // 192
constexpr int KK = 25;             // 5x5 taps

typedef __attribute__((ext_vector_type(2))) float v2f;
typedef __attribute__((ext_vector_type(8))) float v8f;

// ---------------------------------------------------------------------------
// K1: adaptive average pool ctx (B,C,56,56) -> ctx_p (B,C,7,7) ; 8x8 bins
// ---------------------------------------------------------------------------
__global__ __launch_bounds__(256) void pool_kernel(const float* __restrict__ ctx,
                                                   float* __restrict__ ctxp) {
    int i = blockIdx.x * 256 + threadIdx.x;           // over B*C*49
    if (i >= B * C * SS) return;
    int st = i % SS;
    int c  = (i / SS) % C;
    int b  = i / (SS * C);
    int s  = st / S, t = st % S;
    const float* base = ctx + ((size_t)(b * C + c)) * HW + (s * 8) * W + t * 8;
    float sum = 0.f;
#pragma unroll
    for (int r = 0; r < 8; ++r)
#pragma unroll
        for (int q = 0; q < 8; ++q) sum += base[r * W + q];
    ctxp[i] = sum * (1.0f / 64.0f);
}

// ---------------------------------------------------------------------------
// K2: kf[b,d,s] = sum_c Wk[d,c] * ctx_p[b,c,s]       (tiny GEMM, VALU is fine)
// ---------------------------------------------------------------------------
__global__ __launch_bounds__(256) void kf_kernel(const float* __restrict__ Wk,
                                                 const float* __restrict__ ctxp,
                                                 float* __restrict__ kf) {
    int i = blockIdx.x * 256 + threadIdx.x;           // over B*D*49
    if (i >= B * D * SS) return;
    int s = i % SS;
    int d = (i / SS) % D;
    int b = i / (SS * D);
    const float* wr = Wk + (size_t)d * C;
    const float* cp = ctxp + (size_t)b * C * SS + s;
    float acc = 0.f;
    for (int c = 0; c < C; ++c) acc = fmaf(wr[c], cp[c * SS], acc);
    kf[i] = acc;
}

// ---------------------------------------------------------------------------
// K3: M[b,c,s] = sum_d Wq[d,c] * kf[b,d,s]   (fold Wq into kf: scores = x^T M)
//     stored s-padded to 64 (zeros for s>=49)
// ---------------------------------------------------------------------------
__global__ __launch_bounds__(256) void m_kernel(const float* __restrict__ Wq,
                                                const float* __restrict__ kf,
                                                float* __restrict__ M) {
    int i = blockIdx.x * 256 + threadIdx.x;           // over B*C*64
    if (i >= B * C * SP) return;
    int s = i & (SP - 1);
    int c = (i >> 6) % C;
    int b = i / (SP * C);
    float acc = 0.f;
    if (s < SS) {
        const float* kb = kf + (size_t)b * D * SS + s;
        for (int d = 0; d < D; ++d) acc = fmaf(Wq[(size_t)d * C + c], kb[d * SS], acc);
    }
    M[i] = acc;
}

// ---------------------------------------------------------------------------
// K4: fused  scores = x^T M  (WMMA f32 16x16x4)  -> softmax(s) -> dyn = Wwd*A
//     grid: (49 pixel tiles of 64, B), block: 128 (4 waves, 16 pixels each)
// ---------------------------------------------------------------------------
__global__ __launch_bounds__(128) void attn_dyn_kernel(const float* __restrict__ x,
                                                       const float* __restrict__ M,
                                                       const float* __restrict__ Wwd,
                                                       float* __restrict__ dyn) {
    const int b    = blockIdx.y;
    const int pix0 = blockIdx.x * 64;                 // 64 pixels per workgroup
    const int tid  = threadIdx.x;
    const int wave = tid >> 5;
    const int lane = tid & 31;
    const int l16  = lane & 15;
    const int half = lane >> 4;

    __shared__ float xs[32][68];                      // x   K-block (32 ch x 64 px)
    __shared__ float ms[32][68];                      // M   K-block (32 ch x 64 s)
    __shared__ float alds[4][16][68];                 // softmax(A) tile per wave
    __shared__ float wlds[KK][SS];                    // Wwd 25x49

    for (int i = tid; i < KK * SS; i += 128) wlds[i / SS][i % SS] = Wwd[i];

    const float* xb = x + (size_t)b * C * HW;
    const float* mb = M + (size_t)b * C * SP;

    v8f acc[4] = {};                                  // 16 px x 64 s scores

    const int mcol = wave * 16 + l16;                 // this lane's pixel (A row)
    for (int c0 = 0; c0 < C; c0 += 32) {
        __syncthreads();
        for (int i = tid; i < 32 * 64; i += 128) {
            int r = i >> 6, col = i & 63;
            xs[r][col] = xb[(size_t)(c0 + r) * HW + pix0 + col];
            ms[r][col] = mb[(size_t)(c0 + r) * SP + col];
        }
        __syncthreads();
#pragma unroll
        for (int kk = 0; kk < 32; kk += 4) {
            // A-matrix 16x4 f32 layout: lanes 0-15 hold K=0/1, lanes 16-31 K=2/3
            const int krow = kk + 2 * half;
            v2f a;
            a.x = xs[krow + 0][mcol];
            a.y = xs[krow + 1][mcol];
#pragma unroll
            for (int g = 0; g < 4; ++g) {             // 4 N-chunks of 16 s-values
                v2f bf;
                bf.x = ms[krow + 0][g * 16 + l16];
                bf.y = ms[krow + 1][g * 16 + l16];
                acc[g] = __builtin_amdgcn_wmma_f32_16x16x4_f32(
                    false, a, false, bf, (short)0, acc[g], false, false);
            }
        }
    }

    // Softmax over s (64 values per pixel row; s>=49 masked to -inf).
    // C/D layout: VGPR r, lanes 0-15 -> row r, lanes 16-31 -> row r+8;
    // N = g*16 + (lane&15). Reductions stay inside each 16-lane half.
#pragma unroll
    for (int r = 0; r < 8; ++r) {
        float v[4];
        float mx = -1e30f;
#pragma unroll
        for (int g = 0; g < 4; ++g) {
            int s = g * 16 + l16;
            v[g] = (s < SS) ? acc[g][r] : -1e30f;
            mx = fmaxf(mx, v[g]);
        }
#pragma unroll
        for (int m = 8; m >= 1; m >>= 1) mx = fmaxf(mx, __shfl_xor(mx, m, 16));
        float sum = 0.f;
#pragma unroll
        for (int g = 0; g < 4; ++g) { v[g] = __expf(v[g] - mx); sum += v[g]; }
#pragma unroll
        for (int m = 8; m >= 1; m >>= 1) sum += __shfl_xor(sum, m, 16);
        const float inv = 1.0f / sum;
        const int row = r + 8 * half;
#pragma unroll
        for (int g = 0; g < 4; ++g) alds[wave][row][g * 16 + l16] = v[g] * inv;
    }
    __syncthreads();

    // dyn[j, pix] = sum_s Wwd[j,s] * A[pix,s]   (25 x 64 outputs per block)
    float* dynb = dyn + (size_t)b * KK * HW;
    for (int idx = tid; idx < KK * 64; idx += 128) {
        int j = idx >> 6, p = idx & 63;
        const float* arow = alds[p >> 4][p & 15];
        float s = 0.f;
        for (int t = 0; t < SS; ++t) s = fmaf(wlds[j][t], arow[t], s);
        dynb[(size_t)j * HW + pix0 + p] = s;
    }
}

// ---------------------------------------------------------------------------
// K5: out[b,c,h,w] = sum_{dy,dx} x[b,c,h+dy-2,w+dx-2] * dyn[b,dy*5+dx,h,w]
//     14x14 spatial tile, dyn tile in LDS shared across 48 channels per block
// ---------------------------------------------------------------------------
constexpr int TILE = 14;
constexpr int CGRP = 48;

__global__ __launch_bounds__(256) void out_kernel(const float* __restrict__ x,
                                                  const float* __restrict__ dyn,
                                                  float* __restrict__ out) {
    const int b    = blockIdx.z;
    const int cg   = blockIdx.y;                      // 0..7
    const int tile = blockIdx.x;                      // 0..15  (4x4 tiles of 14)
    const int th0  = (tile >> 2) * TILE;
    const int tw0  = (tile & 3) * TILE;
    const int tid  = threadIdx.x;

    __shared__ float dyt[KK][TILE * TILE];            // 25 x 196
    __shared__ float xt[TILE + 4][TILE + 4];          // 18 x 18 halo tile

    for (int i = tid; i < KK * TILE * TILE; i += 256) {
        int j = i / (TILE * TILE), p = i % (TILE * TILE);
        int ph = p / TILE, pw = p % TILE;
        dyt[j][p] = dyn[((size_t)(b * KK + j)) * HW + (th0 + ph) * W + (tw0 + pw)];
    }

    const int p  = tid;
    const int ph = tid / TILE;
    const int pw = tid % TILE;

    for (int ci = 0; ci < CGRP; ++ci) {
        const int c = cg * CGRP + ci;
        const float* xc = x + ((size_t)(b * C + c)) * HW;
        __syncthreads();
        for (int i = tid; i < 18 * 18; i += 256) {
            int r = i / 18, cc = i % 18;
            int gh = th0 - 2 + r, gw = tw0 - 2 + cc;
            xt[r][cc] = (gh >= 0 && gh < H && gw >= 0 && gw < W) ? xc[(size_t)gh * W + gw]
                                                                 : 0.f;
        }
        __syncthreads();
        if (ci + 1 < CGRP) __builtin_prefetch(xc + HW + (size_t)th0 * W, 0, 1);
        if (p < TILE * TILE) {
            float acc = 0.f;
#pragma unroll
            for (int dy = 0; dy < 5; ++dy)
#pragma unroll
                for (int dx = 0; dx < 5; ++dx)
                    acc = fmaf(xt[ph + dy][pw + dx], dyt[dy * 5 + dx][p], acc);
            out[((size_t)(b * C + c)) * HW + (th0 + ph) * W + (tw0 + pw)] = acc;
        }
    }
}

// ---------------------------------------------------------------------------
extern "C" void kernel_launch(void* const* d_in, const int* in_sizes, int n_in,
                              void* d_out, int out_size, void* d_ws, size_t ws_size,
                              hipStream_t stream) {
    const float* x   = (const float*)d_in[0];
    const float* ctx = (const float*)d_in[1];
    const float* Wq  = (const float*)d_in[2];
    const float* Wk  = (const float*)d_in[3];
    const float* Wwd = (const float*)d_in[4];
    float* out = (float*)d_out;

    float* ws   = (float*)d_ws;
    float* ctxp = ws;                                 // B*C*49      =  75,264
    float* kf   = ctxp + (size_t)B * C * SS;          // B*192*49    =  37,632
    float* M    = kf   + (size_t)B * D * SS;          // B*C*64      =  98,304
    float* dyn  = M    + (size_t)B * C * SP;          // B*25*3136   = 313,600
    // total ~525K floats (~2.1 MB) of workspace

    pool_kernel<<<(B * C * SS + 255) / 256, 256, 0, stream>>>(ctx, ctxp);
    kf_kernel<<<(B * D * SS + 255) / 256, 256, 0, stream>>>(Wk, ctxp, kf);
    m_kernel<<<(B * C * SP + 255) / 256, 256, 0, stream>>>(Wq, kf, M);
    attn_dyn_kernel<<<dim3(HW / 64, B), 128, 0, stream>>>(x, M, Wwd, dyn);
    out_kernel<<<dim3(16, C / CGRP, B), 256, 0, stream>>>(x, dyn, out);
}